// MVFNet_89146341196305
// MI455X (gfx1250) — compile-verified
//
#include <hip/hip_runtime.h>
#include <hip/hip_bf16.h>

// CDNA5 / gfx1250. Two-phase memory-bound pipeline (~350 MB total @ 23.3 TB/s):
//   Phase 1: MeanVFE + [Nvox,4]@[4,32] via V_WMMA_F32_16X16X4_F32 + ReLU -> d_ws
//            (voxel_features = 51.2 MB, stays resident in the 192 MB L2)
//   Phase 2: point gather (id==-1 -> 0), NT 128b stores for the 256 MB output.

typedef __attribute__((ext_vector_type(2))) float v2f;
typedef __attribute__((ext_vector_type(4))) float v4f;
typedef __attribute__((ext_vector_type(8))) float v8f;

#define C_IN  4
#define C_OUT 32
#define MAX_PTS 5

// ---------------------------------------------------------------------------
// Phase 1: one wave32 handles a tile of 16 voxels.
//   A-matrix (16x4 f32): lane L (L<16): {mean[v][0], mean[v][1]},
//                        lane L+16:     {mean[v][2], mean[v][3]},  v = base+(L&15)
//   B-matrix (4x16 f32): lane holds col n=(L&15); vgpr0/1 = W rows 2*(L>>4), 2*(L>>4)+1
//   D (v8f): element r -> voxel_features[base + r + 8*(L>>4)][n (+16 for 2nd WMMA)]
// ---------------------------------------------------------------------------
__global__ void mvf_vfe_wmma(const float* __restrict__ voxels,
                             const int*   __restrict__ vnum,
                             const float* __restrict__ W,
                             float*       __restrict__ vfeat,
                             int n_vox) {
    const int gtid  = blockIdx.x * blockDim.x + threadIdx.x;
    const int wave  = gtid >> 5;
    const int lane  = threadIdx.x & 31;
    const int base  = wave * 16;
    if (base >= n_vox) return;               // wave-uniform: EXEC stays all-1s

    const int vrow  = lane & 15;             // voxel slot within tile / column n
    const int khalf = lane >> 4;             // 0 -> channels {0,1}, 1 -> {2,3}

    int v = base + vrow;
    if (v >= n_vox) v = n_vox - 1;           // clamp loads on tail tile

    // ---- mean over MAX_PTS points for this lane's two channels (NT b64 loads) ----
    const v2f* vp = reinterpret_cast<const v2f*>(
        voxels + (size_t)v * (MAX_PTS * C_IN) + khalf * 2);
    v2f s = {0.f, 0.f};
#pragma unroll
    for (int p = 0; p < MAX_PTS; ++p) {
        s += __builtin_nontemporal_load(vp + p * (C_IN / 2));
    }
    int cnt = vnum[v];
    const float cntf = (float)(cnt < 1 ? 1 : cnt);
    v2f a = s / cntf;                        // exact divide, matches reference

    // ---- B operands from W[4][32] (row-major) ----
    const int n = vrow;
    v2f b0, b1;
    b0.x = W[(khalf * 2 + 0) * C_OUT + n];
    b0.y = W[(khalf * 2 + 1) * C_OUT + n];
    b1.x = W[(khalf * 2 + 0) * C_OUT + 16 + n];
    b1.y = W[(khalf * 2 + 1) * C_OUT + 16 + n];

    v8f c0 = {};
    v8f c1 = {};
    // 8 args: (neg_a, A, neg_b, B, c_mod, C, reuse_a, reuse_b)
    c0 = __builtin_amdgcn_wmma_f32_16x16x4_f32(false, a, false, b0, (short)0, c0, false, false);
    c1 = __builtin_amdgcn_wmma_f32_16x16x4_f32(false, a, false, b1, (short)0, c1, false, false);

    // ---- ReLU + store D back to row-major [n_vox, 32] ----
    // Per-lane base pointer; row r is +r*128B, second WMMA half is +64B.
    float* o = vfeat + (size_t)(base + khalf * 8) * C_OUT + n;

    if (base + 16 <= n_vox) {
        // Full tile (the common case): wave-uniform branch, no exec churn,
        // 16 stores with immediate offsets -> clause-friendly.
#pragma unroll
        for (int r = 0; r < 8; ++r) {
            float x0 = c0[r]; x0 = x0 > 0.f ? x0 : 0.f;
            float x1 = c1[r]; x1 = x1 > 0.f ? x1 : 0.f;
            o[r * C_OUT]      = x0;
            o[r * C_OUT + 16] = x1;
        }
    } else {
        // Tail tile: per-lane predicated stores.
#pragma unroll
        for (int r = 0; r < 8; ++r) {
            const int row = base + r + khalf * 8;
            if (row < n_vox) {
                float x0 = c0[r]; x0 = x0 > 0.f ? x0 : 0.f;
                float x1 = c1[r]; x1 = x1 > 0.f ? x1 : 0.f;
                o[r * C_OUT]      = x0;
                o[r * C_OUT + 16] = x1;
            }
        }
    }
}

// ---------------------------------------------------------------------------
// Phase 2: gather voxel_features -> points. 8 lanes per point, 128b each.
// voxel_features (51.2 MB) is L2-resident; output is streamed with NT stores.
// ---------------------------------------------------------------------------
__global__ void mvf_point_gather(const int*   __restrict__ ids,
                                 const float* __restrict__ vfeat,
                                 float*       __restrict__ out,
                                 int n_pts) {
    const int t   = blockIdx.x * blockDim.x + threadIdx.x;
    const int pid = t >> 3;
    const int ch  = (t & 7) * 4;
    if (pid >= n_pts) return;

    const int id = ids[pid];
    v4f val = {0.f, 0.f, 0.f, 0.f};
    if (id >= 0) {
        val = *reinterpret_cast<const v4f*>(vfeat + (size_t)id * C_OUT + ch);
    }
    __builtin_nontemporal_store(val,
        reinterpret_cast<v4f*>(out + (size_t)pid * C_OUT + ch));
}

extern "C" void kernel_launch(void* const* d_in, const int* in_sizes, int n_in,
                              void* d_out, int out_size, void* d_ws, size_t ws_size,
                              hipStream_t stream) {
    const float* voxels = (const float*)d_in[0];   // [n_vox, 5, 4]
    const int*   vnum   = (const int*)  d_in[1];   // [n_vox]
    const int*   ids    = (const int*)  d_in[2];   // [n_pts]
    const float* W      = (const float*)d_in[3];   // [4, 32]
    float*       out    = (float*)d_out;           // [n_pts, 32]

    const int n_vox = in_sizes[1];
    const int n_pts = in_sizes[2];

    float* vfeat = (float*)d_ws;                   // [n_vox, 32] scratch (51.2 MB)
    (void)ws_size; (void)n_in; (void)out_size;

    // Phase 1: one wave per 16 voxels
    {
        const int waves   = (n_vox + 15) / 16;
        const int threads = waves * 32;
        const int block   = 256;
        const int grid    = (threads + block - 1) / block;
        mvf_vfe_wmma<<<grid, block, 0, stream>>>(voxels, vnum, W, vfeat, n_vox);
    }

    // Phase 2: 8 threads per point
    {
        const long long threads = (long long)n_pts * 8;
        const int block = 256;
        const long long grid = (threads + block - 1) / block;
        mvf_point_gather<<<(int)grid, block, 0, stream>>>(ids, vfeat, out, n_pts);
    }
}